// Blocks_86096914416144
// MI455X (gfx1250) — compile-verified
//
#include <hip/hip_runtime.h>
#include <cstdint>
#include <cstddef>

// Problem constants (match reference hyperparams)
#define T_LEN   1024
#define TB      8               // t_len_block (T_LATENCY + 1)
#define NBLK    (T_LEN / TB)    // 128 sequential scan steps
#define BATCH   32
#define N_OUT   1024
#define BN      (BATCH * N_OUT) // 32768 independent (b,n) sequences
#define WG      256             // 8 wave32s per workgroup
#define PD      3               // async prefetch distance (blocks ahead)
#define NBUF    4               // rotating LDS buffers (= PD + 1, power of 2)

typedef __attribute__((address_space(3))) float lds_f32_t;

// gfx1250 async DMA: copy one b32 per lane from global memory into LDS.
// VDST = per-lane LDS byte offset (within WG allocation), VADDR = 64-bit VA.
// Tracked by ASYNCcnt; completes strictly in order.
__device__ __forceinline__ void async_g2l_b32(uint32_t lds_addr, const float* gp) {
    asm volatile("global_load_async_to_lds_b32 %0, %1, off"
                 :
                 : "v"(lds_addr), "v"((uint64_t)(uintptr_t)gp)
                 : "memory");
}
// Release the oldest in-flight group of TB=8 copies, keeping PD groups in flight.
__device__ __forceinline__ void wait_async_le24() {
    asm volatile("s_wait_asynccnt 0x18" ::: "memory");
}

__global__ __launch_bounds__(WG, 1) void snn_block_scan_kernel(
    const float* __restrict__ x,         // [T, B, N]
    const float* __restrict__ beta_raw,  // [N]
    const float* __restrict__ p_raw,     // [N]
    const float* __restrict__ b_raw,     // [N]
    float* __restrict__ out)             // [T, B, N]
{
    // Rotating staging buffers for x: [phase][t][lane]  (32 KB)
    __shared__ float sbuf[NBUF * TB * WG];

    const int tid = threadIdx.x;
    const int bn  = blockIdx.x * WG + tid;   // flat (b,n); consecutive lanes -> consecutive n
    const int n   = bn & (N_OUT - 1);

    // Clamped module "properties"
    const float beta = fminf(fmaxf(beta_raw[n], 0.001f), 0.999f);
    const float p    = fminf(fabsf(p_raw[n]), 0.999f);
    const float bb   = fminf(fmaxf(fabsf(b_raw[n]), 0.001f), 1.0f);
    const float inv_p = 1.0f / p;

    float p_pow[TB + 1];                 // p^0 .. p^8
    p_pow[0] = 1.0f;
#pragma unroll
    for (int t = 1; t <= TB; ++t) p_pow[t] = p_pow[t - 1] * p;

    // Scan carry (per (b,n)): z block, adaptation scalar a, last membrane.
    float z[TB];
#pragma unroll
    for (int t = 0; t < TB; ++t) z[t] = 0.0f;
    float a       = 0.0f;                // a_kernel == p_pow * a
    float int_mem = 0.0f;

    const uint32_t lds0 = (uint32_t)(uintptr_t)(lds_f32_t*)&sbuf[tid];
    // LDS byte address of (phase, t) slot for this lane:
    //   lds0 + (phase*TB + t) * WG * 4

    // Prologue: stage blocks 0..PD-1 into phases 0..PD-1.
#pragma unroll
    for (int k = 0; k < PD; ++k) {
        const float*   g    = x + (size_t)k * TB * BN + bn;
        const uint32_t base = lds0 + (uint32_t)((k * TB) * WG * 4);
#pragma unroll
        for (int t = 0; t < TB; ++t)
            async_g2l_b32(base + (uint32_t)(t * WG * 4), g + (size_t)t * BN);
    }

    for (int blk = 0; blk < NBLK; ++blk) {
        const int cur = blk & (NBUF - 1);

        // Keep the DMA pipeline PD blocks deep. Past the end, clamp the source
        // to the last block: the dummy group lands in the buffer consumed one
        // iteration ago (never re-read), keeping the loop branchless; the
        // implicit wait-idle at s_endpgm drains any leftovers.
        {
            const int      pf  = blk + PD;
            const int      pfs = (pf < NBLK) ? pf : (NBLK - 1);
            const float*   g   = x + (size_t)pfs * TB * BN + bn;
            const uint32_t nb  = lds0 + (uint32_t)(((pf & (NBUF - 1)) * TB) * WG * 4);
#pragma unroll
            for (int t = 0; t < TB; ++t)
                async_g2l_b32(nb + (uint32_t)(t * WG * 4), g + (size_t)t * BN);
        }
        // In-order ASYNCcnt: <=24 outstanding means the oldest 8 (current
        // buffer) have landed. Lanes read only their own slots -> no barrier.
        wait_async_le24();

        float xv[TB];
#pragma unroll
        for (int t = 0; t < TB; ++t) xv[t] = sbuf[(cur * TB + t) * WG + tid];

        // --- previous-block bookkeeping ---
        // spikes sg[t] = (z[t] == 1); maskf = any spike;
        // a_at_spike = a * sum_t p^{t+1} sg[t] + 1/p;
        // decay: p^{#(z[t] > 1)} via select-multiplies (exact integer power).
        float maskf = 0.0f, ssum = 0.0f, pdc = 1.0f;
#pragma unroll
        for (int t = 0; t < TB; ++t) {
            const float sg = (z[t] == 1.0f) ? 1.0f : 0.0f;
            maskf = fmaxf(maskf, sg);
            ssum += sg * p_pow[t + 1];
            pdc  *= (z[t] > 1.0f) ? p : 1.0f;
        }
        const float a_at_spike = a * ssum + inv_p;
        const float new_a      = a_at_spike * pdc;
        a = (maskf > 0.0f) ? new_a : (p_pow[TB] * a);          // a_kernel[-1] = p^8 * a
        const float v_init = (maskf > 0.0f) ? 0.0f : int_mem;  // int_mem * (1 - maskf)

        // --- block forward: recurrences replacing the two einsums ---
        float m = 0.0f, c = 0.0f, zacc = 0.0f;
        float* op = out + (size_t)blk * TB * BN + bn;
#pragma unroll
        for (int t = 0; t < TB; ++t) {
            // refractory masking: xs = (z_prev[t] < maskf) ? 0 : x
            float xs = (z[t] < maskf) ? 0.0f : xv[t];
            const float cur_in = (t == 0) ? (xs + beta * v_init) : xs;
            m = __builtin_fmaf(beta, m, cur_in);                          // causal β-decay sum
            const float vth = __builtin_fmaf(bb * a, p_pow[t + 1], 1.0f); // 1 + bb * a_k[t]
            const float f   = (m > vth) ? 1.0f : 0.0f;                    // Heaviside fwd
            c    += f;                                                    // phi kernel as
            zacc += c;                                                    // double prefix sum
            z[t] = zacc;                                                  // carry z (old z[t] already consumed)
            // Streaming, never re-read: non-temporal store keeps L2 clean.
            __builtin_nontemporal_store((zacc == 1.0f) ? 1.0f : 0.0f,
                                        op + (size_t)t * BN);
        }
        int_mem = m;                                                      // membrane[-1]
    }
}

extern "C" void kernel_launch(void* const* d_in, const int* in_sizes, int n_in,
                              void* d_out, int out_size, void* d_ws, size_t ws_size,
                              hipStream_t stream) {
    (void)in_sizes; (void)n_in; (void)out_size; (void)d_ws; (void)ws_size;
    const float* x        = (const float*)d_in[0];
    const float* beta_raw = (const float*)d_in[1];
    const float* p_raw    = (const float*)d_in[2];
    const float* b_raw    = (const float*)d_in[3];
    float* out            = (float*)d_out;

    dim3 grid(BN / WG);   // 128 workgroups x 256 threads = 32768 lanes (1024 wave32s)
    dim3 block(WG);
    hipLaunchKernelGGL(snn_block_scan_kernel, grid, block, 0, stream,
                       x, beta_raw, p_raw, b_raw, out);
}